// DoGLaplacian_44040594653397
// MI455X (gfx1250) — compile-verified
//
#include <hip/hip_runtime.h>

typedef float v2f __attribute__((ext_vector_type(2)));
typedef float v8f __attribute__((ext_vector_type(8)));

#define DIM   160
#define TILE  16
#define HALO  7                         // radius of the sigma=1.6 kernel
#define RAWE  (TILE + 2*HALO)           // 30
#define RAWN  (RAWE*RAWE*RAWE)          // 27000 floats
#define NLINES_X (RAWE*RAWE)            // 900 (y,z) lines in phase X
#define LSTR  916                       // i1 line stride: mult of 4 (b128 align), 916%64=20 -> conflict-free
#define I1N   (TILE*LSTR)               // 14656 floats (per sigma), padded to 912+ lines
#define YSTR  20                        // i2 y-row stride: 20%64 -> 16 distinct banks, mult of 4
#define ZSTR  (TILE*YSTR)               // 320 floats per z-plane
#define I2N   (RAWE*ZSTR)               // 9600 floats (per sigma)
#define SMEM_FLOATS (RAWN + 2*I1N)      // 56312 -> 225248 bytes

#define KL 9
#define RL 4
#define KH 15
#define RH 7
#define SIGL 1.0f
#define SIGH 1.6f

// D = A(16x4) * B(4x16) + C, fp32 WMMA (wave32). 8-arg VOP3P form.
__device__ __forceinline__ v8f wmma_f32_16x16x4(v2f a, v2f b, v8f c) {
  return __builtin_amdgcn_wmma_f32_16x16x4_f32(
      /*neg_a=*/false, a, /*neg_b=*/false, b,
      /*c_mod=*/(short)0, c, /*reuse_a=*/false, /*reuse_b=*/false);
}

__device__ __forceinline__ float gauss_tap(int d, int K, int R, float inv2s2, float invn) {
  if ((unsigned)d >= (unsigned)K) return 0.0f;
  float t = (float)(d - R);
  return __expf(-t * t * inv2s2) * invn;
}

__global__ __launch_bounds__(256, 1)
void dog3d_wmma_kernel(const float* __restrict__ in, float* __restrict__ out)
{
  extern __shared__ float smem[];
  float* raw = smem;                    // phase-1 input tile, 27000 floats
  float* i1l = smem + RAWN;             // after-X, low:  [x][line], line-stride LSTR
  float* i1h = smem + RAWN + I1N;       // after-X, high
  float* i2l = smem;                    // after-Y, low:  [z][y][x], strides ZSTR/YSTR/1
  float* i2h = smem + I2N;              // (reuses raw region; 2*9600 <= 27000)

  const int tid  = threadIdx.x;
  const int lane = tid & 31;
  const int wave = tid >> 5;

  const int x0 = blockIdx.x * TILE;
  const int y0 = blockIdx.y * TILE;
  const int zt = blockIdx.z % 10;
  const int b  = blockIdx.z / 10;
  const int z0 = zt * TILE;

  // ---------------- constant Toeplitz B fragments (shared by all 3 axes) ---
  const int n    = lane & 15;           // output-column index this lane holds
  const int koff = (lane & 16) >> 3;    // K sub-offset in {0,2} per ISA A/B layout

  const float inv2s2_l = 0.5f / (SIGL * SIGL);
  const float inv2s2_h = 0.5f / (SIGH * SIGH);
  float suml = 0.0f, sumh = 0.0f;
#pragma unroll
  for (int d = 0; d < KL; ++d) { float t = (float)(d - RL); suml += __expf(-t*t*inv2s2_l); }
#pragma unroll
  for (int d = 0; d < KH; ++d) { float t = (float)(d - RH); sumh += __expf(-t*t*inv2s2_h); }
  const float invl = 1.0f / suml;
  const float invh = 1.0f / sumh;

  v2f bl[6];                            // low: K = 24 = 6 chunks of 4
#pragma unroll
  for (int c = 0; c < 6; ++c) {
    const int i0 = 4*c + koff;
    bl[c].x = gauss_tap(i0     - n, KL, RL, inv2s2_l, invl);
    bl[c].y = gauss_tap(i0 + 1 - n, KL, RL, inv2s2_l, invl);
  }
  v2f bh[8];                            // high: K = 30 -> 32 = 8 chunks (pad rows are 0)
#pragma unroll
  for (int c = 0; c < 8; ++c) {
    const int i0 = 4*c + koff;
    bh[c].x = gauss_tap(i0     - n, KH, RH, inv2s2_h, invh);
    bh[c].y = gauss_tap(i0 + 1 - n, KH, RH, inv2s2_h, invh);
  }

  // ---------------- load 30^3 haloed tile with replicate borders -----------
  for (int r = tid; r < RAWN; r += 256) {
    const int xi = r % RAWE;
    const int t2 = r / RAWE;
    const int yi = t2 % RAWE;
    const int zi = t2 / RAWE;
    const int gx = min(max(x0 - HALO + xi, 0), DIM - 1);
    const int gy = min(max(y0 - HALO + yi, 0), DIM - 1);
    const int gz = min(max(z0 - HALO + zi, 0), DIM - 1);
    raw[r] = in[(((size_t)b * DIM + gz) * DIM + gy) * DIM + gx];
  }
  __syncthreads();

  // ---------------- phase X: conv along x for all 900 (y,z) lines ----------
  // line l = zi*30 + yi ; 57 groups of 16 lines. A rows of the tail group are
  // clamped (finite data), results land in the padded i1 region (never read).
  for (int g = wave; g < 57; g += 8) {
    const int lbase = g * 16;
    const int lA = min(lbase + (lane & 15), NLINES_X - 1);
    v8f accl = {}; v8f acch = {};
#pragma unroll
    for (int c = 0; c < 6; ++c) {               // low: input x in [-4,19] -> raw xi = i+3
      const int xi = 3 + 4*c + koff;
      v2f a; a.x = raw[lA * RAWE + xi]; a.y = raw[lA * RAWE + xi + 1];
      accl = wmma_f32_16x16x4(a, bl[c], accl);
    }
#pragma unroll
    for (int c = 0; c < 8; ++c) {               // high: input x in [-7,22] -> raw xi = i
      const int xi = 4*c + koff;
      const int xa = (c == 7) ? min(xi,     RAWE - 1) : xi;      // pad rows hit B==0
      const int xb = (c == 7) ? min(xi + 1, RAWE - 1) : xi + 1;
      v2f a; a.x = raw[lA * RAWE + xa]; a.y = raw[lA * RAWE + xb];
      acch = wmma_f32_16x16x4(a, bh[c], acch);
    }
    const int xout = lane & 15;
    const int base = xout * LSTR + lbase + ((lane & 16) >> 1);   // + mhi(0|8)
#pragma unroll
    for (int v = 0; v < 8; ++v) {               // 8 consecutive floats -> 2x ds_store_b128
      i1l[base + v] = accl[v];
      i1h[base + v] = acch[v];
    }
  }
  __syncthreads();

  // ---------------- phase Y: conv along y; one group per zi (30 groups) ----
  for (int g = wave; g < RAWE; g += 8) {        // g = zi
    const int x = lane & 15;                    // A row (line) = x
    v8f accl = {}; v8f acch = {};
#pragma unroll
    for (int c = 0; c < 6; ++c) {
      const int yi = 3 + 4*c + koff;
      v2f a; a.x = i1l[x * LSTR + g * RAWE + yi];
             a.y = i1l[x * LSTR + g * RAWE + yi + 1];
      accl = wmma_f32_16x16x4(a, bl[c], accl);
    }
#pragma unroll
    for (int c = 0; c < 8; ++c) {
      const int yi = 4*c + koff;
      const int ya = (c == 7) ? min(yi,     RAWE - 1) : yi;
      const int yb = (c == 7) ? min(yi + 1, RAWE - 1) : yi + 1;
      v2f a; a.x = i1h[x * LSTR + g * RAWE + ya];
             a.y = i1h[x * LSTR + g * RAWE + yb];
      acch = wmma_f32_16x16x4(a, bh[c], acch);
    }
    const int yout = lane & 15;
    const int base = g * ZSTR + yout * YSTR + ((lane & 16) >> 1); // + xm_hi(0|8)
#pragma unroll
    for (int v = 0; v < 8; ++v) {               // 8 consecutive floats -> 2x ds_store_b128
      i2l[base + v] = accl[v];
      i2h[base + v] = acch[v];
    }
  }
  __syncthreads();

  // ---------------- phase Z: conv along z; one group per y (16 groups) -----
  for (int g = wave; g < TILE; g += 8) {        // g = y
    const int x = lane & 15;                    // A row (line) = x
    v8f accl = {}; v8f acch = {};
#pragma unroll
    for (int c = 0; c < 6; ++c) {
      const int zi = 3 + 4*c + koff;
      v2f a; a.x = i2l[zi       * ZSTR + g * YSTR + x];
             a.y = i2l[(zi + 1) * ZSTR + g * YSTR + x];
      accl = wmma_f32_16x16x4(a, bl[c], accl);
    }
#pragma unroll
    for (int c = 0; c < 8; ++c) {
      const int zi = 4*c + koff;
      const int za = (c == 7) ? min(zi,     RAWE - 1) : zi;
      const int zb = (c == 7) ? min(zi + 1, RAWE - 1) : zi + 1;
      v2f a; a.x = i2h[za * ZSTR + g * YSTR + x];
             a.y = i2h[zb * ZSTR + g * YSTR + x];
      acch = wmma_f32_16x16x4(a, bh[c], acch);
    }
    const int zo = lane & 15;
    const int xmb = (lane & 16) >> 1;           // 0 or 8
    const size_t o = (((size_t)b * DIM + (z0 + zo)) * DIM + (y0 + g)) * DIM + (x0 + xmb);
#pragma unroll
    for (int v = 0; v < 8; ++v)                 // 8 consecutive floats -> 2x global_store_b128
      out[o + v] = accl[v] - acch[v];           // DoG = low - high
  }
}

extern "C" void kernel_launch(void* const* d_in, const int* in_sizes, int n_in,
                              void* d_out, int out_size, void* d_ws, size_t ws_size,
                              hipStream_t stream) {
  (void)in_sizes; (void)n_in; (void)d_ws; (void)ws_size; (void)out_size;
  const float* x = (const float*)d_in[0];
  float* out = (float*)d_out;

  const size_t shmem = (size_t)SMEM_FLOATS * sizeof(float);   // 225248 B < 320 KB/WGP
  // Opt in to >64KB dynamic LDS (deterministic, non-stream call).
  (void)hipFuncSetAttribute((const void*)dog3d_wmma_kernel,
                            hipFuncAttributeMaxDynamicSharedMemorySize, (int)shmem);

  dim3 grid(DIM / TILE, DIM / TILE, (DIM / TILE) * 2);        // 10 x 10 x 20
  dog3d_wmma_kernel<<<grid, dim3(256), shmem, stream>>>(x, out);
}